// SurfaceNet_62637803044905
// MI455X (gfx1250) — compile-verified
//
#include <hip/hip_runtime.h>
#include <hip/hip_bf16.h>

#define N_NODES 100000
#define N_EDGES 1600000

typedef float v2f __attribute__((ext_vector_type(2)));
typedef float v8f __attribute__((ext_vector_type(8)));

// ---------------------------------------------------------------- utilities
__global__ void zero_f32(float* __restrict__ p, long n) {
    long i = (long)blockIdx.x * blockDim.x + threadIdx.x;
    long stride = (long)gridDim.x * blockDim.x;
    for (; i < n; i += stride) p[i] = 0.0f;
}

__global__ void degree_kernel(const int* __restrict__ dst, float* __restrict__ deg, int E) {
    int e = blockIdx.x * blockDim.x + threadIdx.x;
    if (e < E) unsafeAtomicAdd(&deg[dst[e]], 1.0f);
}

__global__ void invdeg_kernel(const float* __restrict__ deg, float* __restrict__ inv, int n) {
    int i = blockIdx.x * blockDim.x + threadIdx.x;
    if (i < n) {
        float d = deg[i];
        inv[i] = (d > 0.0f) ? (1.0f / fmaxf(d, 1.0f)) : 0.0f;
    }
}

// ------------------------------------------------- edge gather + scatter-add
// one thread = one (edge, 4-float chunk); consecutive lanes cover consecutive
// chunks of the same edge -> coalesced b128 gather + global_atomic_add_f32.
__global__ void scatter_add_kernel(const float* __restrict__ x,
                                   const int* __restrict__ src,
                                   const int* __restrict__ dst,
                                   float* __restrict__ agg,
                                   int E, int F, int chunkShift) {
    long total   = (long)E << chunkShift;          // E * (F/4)
    long idx     = (long)blockIdx.x * blockDim.x + threadIdx.x;
    long stride  = (long)gridDim.x * blockDim.x;
    int  cmask   = (1 << chunkShift) - 1;
    for (; idx < total; idx += stride) {
        int e  = (int)(idx >> chunkShift);
        int f4 = (int)(idx & cmask) << 2;
        int s  = src[e];
        int d  = dst[e];
        const float4 v = *(const float4*)(x + (size_t)s * F + f4);
        float* ap = agg + (size_t)d * F + f4;
        unsafeAtomicAdd(ap + 0, v.x);
        unsafeAtomicAdd(ap + 1, v.y);
        unsafeAtomicAdd(ap + 2, v.z);
        unsafeAtomicAdd(ap + 3, v.w);
    }
}

// ------------------------------------------------------------- WMMA GEMM
// out[n, dout] = act( (scale?A1*scale:A1) @ W1^T  (+ A2 @ W2^T)  + bias )
// One wave computes one 16x16 output tile with V_WMMA_F32_16X16X4_F32,
// iterating K in steps of 4. fp32 A layout: m = lane%16, k-pair = (lane/16)*2.
// fp32 B layout (K x N = W^T): n = lane%16, same k-pair split.
// C/D layout: VGPR r -> row r (lanes 0-15) / r+8 (lanes 16-31), col = lane%16.
__global__ __launch_bounds__(128)
void gemm_wmma_kernel(const float* __restrict__ A1, const float* __restrict__ scale,
                      const float* __restrict__ W1,
                      const float* __restrict__ A2, const float* __restrict__ W2,
                      const float* __restrict__ bias, float* __restrict__ out,
                      int Nnodes, int K, int dout, int relu) {
    int wave = threadIdx.x >> 5;
    int lane = threadIdx.x & 31;
    int m0 = (blockIdx.x * 4 + wave) * 16;
    if (m0 >= Nnodes) return;                 // wave-uniform (Nnodes % 16 == 0)
    int n0    = blockIdx.y * 16;
    int mrow  = lane & 15;
    int khalf = (lane >> 4) << 1;             // 0 or 2
    int node  = m0 + mrow;
    int ncol  = n0 + mrow;
    int wrow  = (ncol < dout) ? ncol : 0;     // clamp; masked out at store

    v8f acc = {};
    float sc = scale ? scale[node] : 1.0f;

    const float* ap = A1 + (size_t)node * K + khalf;
    const float* wp = W1 + (size_t)wrow * K + khalf;
    for (int k = 0; k < K; k += 4) {
        v2f a, b;
        a.x = ap[k] * sc;  a.y = ap[k + 1] * sc;
        b.x = wp[k];       b.y = wp[k + 1];
        acc = __builtin_amdgcn_wmma_f32_16x16x4_f32(false, a, false, b,
                                                    (short)0, acc, false, false);
    }
    if (A2) {
        const float* ap2 = A2 + (size_t)node * K + khalf;
        const float* wp2 = W2 + (size_t)wrow * K + khalf;
        for (int k = 0; k < K; k += 4) {
            v2f a, b;
            a.x = ap2[k];  a.y = ap2[k + 1];
            b.x = wp2[k];  b.y = wp2[k + 1];
            acc = __builtin_amdgcn_wmma_f32_16x16x4_f32(false, a, false, b,
                                                        (short)0, acc, false, false);
        }
    }
    if (ncol < dout) {
        float bv   = bias[ncol];
        int  rbase = m0 + ((lane >> 4) << 3);
        #pragma unroll
        for (int r = 0; r < 8; ++r) {
            float v = acc[r] + bv;
            if (relu) v = fmaxf(v, 0.0f);
            out[(size_t)(rbase + r) * dout + ncol] = v;
        }
    }
}

// ---------------------------------------------------------------- launcher
static inline int cdiv(long a, long b) { return (int)((a + b - 1) / b); }

extern "C" void kernel_launch(void* const* d_in, const int* in_sizes, int n_in,
                              void* d_out, int out_size, void* d_ws, size_t ws_size,
                              hipStream_t stream) {
    const float* x    = (const float*)d_in[0];
    const int*   edge = (const int*)d_in[1];
    const int*   src  = edge;
    const int*   dst  = edge + N_EDGES;
    const float* Wl0 = (const float*)d_in[2],  *bl0 = (const float*)d_in[3],  *Wr0 = (const float*)d_in[4];
    const float* Wl1 = (const float*)d_in[5],  *bl1 = (const float*)d_in[6],  *Wr1 = (const float*)d_in[7];
    const float* Wl2 = (const float*)d_in[8],  *bl2 = (const float*)d_in[9],  *Wr2 = (const float*)d_in[10];
    const float* Wd1 = (const float*)d_in[11], *bd1 = (const float*)d_in[12];
    const float* Wd2 = (const float*)d_in[13], *bd2 = (const float*)d_in[14];

    float* ws   = (float*)d_ws;
    float* deg  = ws;                              // N_NODES
    float* inv  = deg  + N_NODES;                  // N_NODES
    float* agg  = inv  + N_NODES;                  // N_NODES*128
    float* bufA = agg  + (size_t)N_NODES * 128;    // N_NODES*128
    float* bufB = bufA + (size_t)N_NODES * 128;    // N_NODES*128

    const int BT = 256;
    const int gm = cdiv(N_NODES, 64);              // node tiles / 4 waves

    // degrees
    zero_f32<<<cdiv(N_NODES, BT), BT, 0, stream>>>(deg, N_NODES);
    degree_kernel<<<cdiv(N_EDGES, BT), BT, 0, stream>>>(dst, deg, N_EDGES);
    invdeg_kernel<<<cdiv(N_NODES, BT), BT, 0, stream>>>(deg, inv, N_NODES);

    // ---- layer 0: F 64 -> 128, ReLU
    zero_f32<<<cdiv((long)N_NODES * 64, BT), BT, 0, stream>>>(agg, (long)N_NODES * 64);
    scatter_add_kernel<<<cdiv((long)N_EDGES * 16, BT), BT, 0, stream>>>(x, src, dst, agg, N_EDGES, 64, 4);
    gemm_wmma_kernel<<<dim3(gm, 8), 128, 0, stream>>>(agg, inv, Wl0, x, Wr0, bl0, bufA,
                                                      N_NODES, 64, 128, 1);

    // ---- layer 1: F 128 -> 128, ReLU
    zero_f32<<<cdiv((long)N_NODES * 128, BT), BT, 0, stream>>>(agg, (long)N_NODES * 128);
    scatter_add_kernel<<<cdiv((long)N_EDGES * 32, BT), BT, 0, stream>>>(bufA, src, dst, agg, N_EDGES, 128, 5);
    gemm_wmma_kernel<<<dim3(gm, 8), 128, 0, stream>>>(agg, inv, Wl1, bufA, Wr1, bl1, bufB,
                                                      N_NODES, 128, 128, 1);

    // ---- layer 2: F 128 -> 64, no conv ReLU but decoder ReLU fused here
    zero_f32<<<cdiv((long)N_NODES * 128, BT), BT, 0, stream>>>(agg, (long)N_NODES * 128);
    scatter_add_kernel<<<cdiv((long)N_EDGES * 32, BT), BT, 0, stream>>>(bufB, src, dst, agg, N_EDGES, 128, 5);
    gemm_wmma_kernel<<<dim3(gm, 4), 128, 0, stream>>>(agg, inv, Wl2, bufB, Wr2, bl2, bufA,
                                                      N_NODES, 128, 64, 1);   // bufA = relu(h2) [N,64]

    // ---- decoder: 64 -> 128 (ReLU) -> 2
    gemm_wmma_kernel<<<dim3(gm, 8), 128, 0, stream>>>(bufA, nullptr, Wd1, nullptr, nullptr, bd1, bufB,
                                                      N_NODES, 64, 128, 1);   // bufB = hd [N,128]
    gemm_wmma_kernel<<<dim3(gm, 1), 128, 0, stream>>>(bufB, nullptr, Wd2, nullptr, nullptr, bd2,
                                                      (float*)d_out, N_NODES, 128, 2, 0);
}